// MNISTKANNetwork_41987600286128
// MI455X (gfx1250) — compile-verified
//
#include <hip/hip_runtime.h>

typedef __attribute__((ext_vector_type(2))) float v2f;
typedef __attribute__((ext_vector_type(8))) float v8f;

#define KORD 3
#define NKNOT 12            // GRID_NUM + 2K + 1 knots, uniform -2.2..2.2 step 0.4
#define NB    8             // GRID_NUM + K basis functions
#define NF1   20            // 2 silu + 16 basis + 2 zero pad (K multiple of 4)
#define ROWS_PER_BLOCK 256
#define HSTRIDE 9           // padded LDS stride for h tile (bank-conflict dodge)

__device__ __forceinline__ float knot(int t) { return -2.2f + 0.4f * (float)t; }

__device__ __forceinline__ float siluf(float v) {
    return v / (1.0f + __expf(-v));
}

// Degree-3 Cox-de-Boor recursion on the uniform extended grid, matches reference.
__device__ __forceinline__ void bspline8(float x, float bs[NB]) {
    float B[NKNOT - 1];
#pragma unroll
    for (int j = 0; j < NKNOT - 1; ++j)
        B[j] = (x >= knot(j) && x < knot(j + 1)) ? 1.0f : 0.0f;
#pragma unroll
    for (int p = 1; p <= KORD; ++p) {
        const float inv = 1.0f / (0.4f * (float)p);   // constant after unroll
#pragma unroll
        for (int j = 0; j < NKNOT - 1 - p; ++j) {     // in-place L->R is safe
            float left  = (x - knot(j)) * inv;
            float right = (knot(j + p + 1) - x) * inv;
            B[j] = left * B[j] + right * B[j + 1];
        }
    }
#pragma unroll
    for (int j = 0; j < NB; ++j) bs[j] = B[j];
}

// Fused layer-1 weight matrix W1[k][n], k in [0,20), n in [0,16) (n>=5 -> 0).
__device__ __forceinline__ float w1_elem(int k, int n,
                                         const float* __restrict__ coef1,
                                         const float* __restrict__ sb1,
                                         const float* __restrict__ sp1) {
    if (n >= 5) return 0.0f;
    if (k == 0) return sb1[n];                                   // i=0 silu row
    if (k == 1) return sb1[5 + n];                               // i=1 silu row
    if (k < 2 + NB)     return sp1[n]     * coef1[n * NB + (k - 2)];            // i=0
    if (k < 2 + 2 * NB) return sp1[5 + n] * coef1[(5 + n) * NB + (k - 2 - NB)]; // i=1
    return 0.0f;                                                 // K pad
}

__global__ __launch_bounds__(ROWS_PER_BLOCK) void kan_fused_kernel(
    const float* __restrict__ x,
    const float* __restrict__ coef1, const float* __restrict__ sb1,
    const float* __restrict__ sp1,
    const float* __restrict__ coef2, const float* __restrict__ sb2,
    const float* __restrict__ sp2,
    const float* __restrict__ fcw,   const float* __restrict__ fcb,
    float* __restrict__ out, int nrows)
{
    __shared__ float lds_feat[ROWS_PER_BLOCK * NF1];     // 20 KB
    __shared__ float lds_h[ROWS_PER_BLOCK * HSTRIDE];    //  9 KB

    const int t    = threadIdx.x;
    const int lane = t & 31;
    const int wave = t >> 5;
    const int blockBase = blockIdx.x * ROWS_PER_BLOCK;
    const int row  = blockBase + t;

    __builtin_prefetch(&x[(size_t)blockBase * 2], 0, 1);   // global_prefetch_b8

    // ---- phase 1: per-row feature vector (VALU) -> LDS ----
    float x0 = 0.0f, x1 = 0.0f;
    if (row < nrows) {
        x0 = x[(size_t)row * 2 + 0];
        x1 = x[(size_t)row * 2 + 1];
    }
    float bsa[NB], bsb[NB];
    bspline8(x0, bsa);
    bspline8(x1, bsb);
    {
        float* f = &lds_feat[t * NF1];
        const bool live = (row < nrows);
        f[0] = live ? siluf(x0) : 0.0f;
        f[1] = live ? siluf(x1) : 0.0f;
#pragma unroll
        for (int k = 0; k < NB; ++k) {
            f[2 + k]      = live ? bsa[k] : 0.0f;
            f[2 + NB + k] = live ? bsb[k] : 0.0f;
        }
        f[18] = 0.0f; f[19] = 0.0f;
    }

    // ---- layer-1 weight B-fragments, WMMA B layout (built once, uniform flow) ----
    const int n  = lane & 15;    // N column (C/D) / M row (A)
    const int hh = lane >> 4;    // lane-half selects K pair
    v2f bfrag[5];
#pragma unroll
    for (int j = 0; j < 5; ++j) {
        int kb = 4 * j + 2 * hh;
        bfrag[j].x = w1_elem(kb,     n, coef1, sb1, sp1);
        bfrag[j].y = w1_elem(kb + 1, n, coef1, sb1, sp1);
    }

    __syncthreads();

    // ---- phase 2: V_WMMA_F32_16X16X4_F32, two 16-row tiles per wave, K=20 ----
    const int R0 = wave * 32;
    const int R1 = R0 + 16;
    v8f acc0 = {}, acc1 = {};
#pragma unroll
    for (int j = 0; j < 5; ++j) {
        int kb = 4 * j + 2 * hh;
        // A layout: M = lane&15, lanes 0-15 -> K={kb, kb+1}, lanes 16-31 -> K={4j+2,4j+3}
        v2f a0 = *(const v2f*)&lds_feat[(R0 + n) * NF1 + kb];   // 8B aligned (kb even)
        v2f a1 = *(const v2f*)&lds_feat[(R1 + n) * NF1 + kb];
        acc0 = __builtin_amdgcn_wmma_f32_16x16x4_f32(
            false, a0, false, bfrag[j], (short)0, acc0, false, false);
        acc1 = __builtin_amdgcn_wmma_f32_16x16x4_f32(
            false, a1, false, bfrag[j], (short)0, acc1, false, false);
    }

    // C/D layout: lane holds h[M = v + 8*hh][N = n] in acc[v]; keep N<5
    if (n < 5) {
#pragma unroll
        for (int v = 0; v < 8; ++v) {
            lds_h[(R0 + v + 8 * hh) * HSTRIDE + n] = acc0[v];
            lds_h[(R1 + v + 8 * hh) * HSTRIDE + n] = acc1[v];
        }
    }

    __syncthreads();

    // ---- phase 3: layer 2 (5->1) + fc, scalar per row (N=1: WMMA pointless) ----
    if (row < nrows) {
        float a = 0.0f;
#pragma unroll
        for (int d = 0; d < 5; ++d) {
            float hv = lds_h[t * HSTRIDE + d];
            float bs2[NB];
            bspline8(hv, bs2);
            float s = 0.0f;
#pragma unroll
            for (int k = 0; k < NB; ++k) s += coef2[d * NB + k] * bs2[k];
            a += sb2[d] * siluf(hv) + sp2[d] * s;
        }
        out[row] = a * fcw[0] + fcb[0];
    }
}

extern "C" void kernel_launch(void* const* d_in, const int* in_sizes, int n_in,
                              void* d_out, int out_size, void* d_ws, size_t ws_size,
                              hipStream_t stream) {
    (void)n_in; (void)d_ws; (void)ws_size; (void)out_size;
    const float* x     = (const float*)d_in[0];
    const float* coef1 = (const float*)d_in[1];
    const float* sb1   = (const float*)d_in[2];
    const float* sp1   = (const float*)d_in[3];
    const float* coef2 = (const float*)d_in[4];
    const float* sb2   = (const float*)d_in[5];
    const float* sp2   = (const float*)d_in[6];
    const float* fcw   = (const float*)d_in[7];
    const float* fcb   = (const float*)d_in[8];
    float* out = (float*)d_out;

    const int nrows   = in_sizes[0] / 2;
    const int nblocks = (nrows + ROWS_PER_BLOCK - 1) / ROWS_PER_BLOCK;

    kan_fused_kernel<<<nblocks, ROWS_PER_BLOCK, 0, stream>>>(
        x, coef1, sb1, sp1, coef2, sb2, sp2, fcw, fcb, out, nrows);
}